// ConnectedFilterLayerByThresholds_25984552141120
// MI455X (gfx1250) — compile-verified
//
#include <hip/hip_runtime.h>
#include <math.h>

// Problem constants (from the reference file).
#define BC   12
#define GG   2
#define NN   500000
#define HH   512
#define WW   512
#define HWP  (HH * WW)          // 262144 pixels per batch
#define BETA_F  1000.0f
#define CLAMP_V 12.0f

// 8-byte packed node record: one global_load_b64 per chase step.
struct __attribute__((aligned(8))) NodePack {
    float contrib;
    int   parent;
};

__device__ __forceinline__ float node_contrib(float a0, float a1, float t0, float t1,
                                              float lev, float lev_par, int i) {
    float d = fabsf(a0 - t0) * fabsf(a1 - t1);
    float x = -BETA_F * d;
    x = fminf(fmaxf(x, -CLAMP_V), CLAMP_V);        // matches jnp.clip
    float sig = 1.0f / (1.0f + expf(-x));          // sigmoid, x in [-12, 0]
    float res = (i == 0) ? lev : (lev - lev_par);  // res[:,0] = levels[:,0]
    return sig * res;
}

// Kernel 1: streaming pass -> packed {contrib, parent} per node.
// attrs is read exactly once -> non-temporal loads keep it out of L2 so the
// packed array + parent gathers stay resident for the chase kernel.
__global__ void build_packed_kernel(const float* __restrict__ attrs,
                                    const float* __restrict__ thr,
                                    const float* __restrict__ levels,
                                    const int*   __restrict__ parent,
                                    NodePack*    __restrict__ pack) {
    const int i = blockIdx.x * blockDim.x + threadIdx.x;
    const int b = blockIdx.y;
    if (i >= NN) return;

    const float t0 = thr[0];
    const float t1 = thr[1];

    const long long abase = (long long)b * GG * NN;
    const long long nbase = (long long)b * NN;

    float a0 = __builtin_nontemporal_load(&attrs[abase + i]);
    float a1 = __builtin_nontemporal_load(&attrs[abase + NN + i]);

    int   p   = parent[nbase + i];
    float lev = levels[nbase + i];
    float lp  = (i == 0) ? lev : levels[nbase + p];

    NodePack np;
    np.contrib = node_contrib(a0, a1, t0, t1, lev, lp, i);
    np.parent  = (i == 0) ? 0 : p;
    pack[nbase + i] = np;   // single b64 store, L2-resident (RT)
}

// Kernel 2: per-pixel root-path chase over the packed array.
// Parents strictly decrease, so the walk terminates at node 0; a hard cap
// guarantees no hang even with pathological inputs.
__global__ void walk_packed_kernel(const NodePack* __restrict__ pack,
                                   const int*      __restrict__ pix,
                                   float*          __restrict__ out) {
    const int p = blockIdx.x * blockDim.x + threadIdx.x;   // pixel within batch
    const int b = blockIdx.y;
    if (p >= HWP) return;

    const long long gid = (long long)b * HWP + p;
    int node = pix[gid];

    const unsigned long long* __restrict__ pb =
        (const unsigned long long*)(pack + (long long)b * NN);

    float acc = 0.0f;
    #pragma unroll 1
    for (int it = 0; it <= NN; ++it) {
        unsigned long long raw = pb[node];                 // global_load_b64 (L2 hit)
        NodePack np = *(const NodePack*)&raw;
        acc += np.contrib;
        if (node == 0) break;
        node = np.parent;
    }
    __builtin_nontemporal_store(acc, &out[gid]);           // streamed output, TH=NT
}

// Fallback: no-scratch direct walk (recomputes contrib inline each step).
// All gathered arrays (attrs 4MB + levels 2MB + parent 2MB per batch) still
// fit in the 192MB L2 across 12 batches.
__global__ void walk_direct_kernel(const float* __restrict__ attrs,
                                   const float* __restrict__ thr,
                                   const float* __restrict__ levels,
                                   const int*   __restrict__ parent,
                                   const int*   __restrict__ pix,
                                   float*       __restrict__ out) {
    const int p = blockIdx.x * blockDim.x + threadIdx.x;
    const int b = blockIdx.y;
    if (p >= HWP) return;

    const long long gid   = (long long)b * HWP + p;
    const long long abase = (long long)b * GG * NN;
    const long long nbase = (long long)b * NN;

    const float t0 = thr[0];
    const float t1 = thr[1];

    int node = pix[gid];
    float acc = 0.0f;

    #pragma unroll 1
    for (int it = 0; it <= NN; ++it) {
        float a0  = attrs[abase + node];
        float a1  = attrs[abase + NN + node];
        int   par = parent[nbase + node];
        float lev = levels[nbase + node];
        float lp  = (node == 0) ? lev : levels[nbase + par];
        acc += node_contrib(a0, a1, t0, t1, lev, lp, node);
        if (node == 0) break;
        node = par;
    }
    __builtin_nontemporal_store(acc, &out[gid]);
}

extern "C" void kernel_launch(void* const* d_in, const int* in_sizes, int n_in,
                              void* d_out, int out_size, void* d_ws, size_t ws_size,
                              hipStream_t stream) {
    (void)in_sizes; (void)n_in; (void)out_size;

    const float* attrs  = (const float*)d_in[0];
    const float* thr    = (const float*)d_in[1];
    const float* levels = (const float*)d_in[2];
    const int*   parent = (const int*)  d_in[3];
    const int*   pix    = (const int*)  d_in[4];
    float*       out    = (float*)d_out;

    const size_t need = (size_t)BC * NN * sizeof(NodePack);   // 48 MB

    const int threads = 256;                                  // 8 wave32 waves
    dim3 gridNodes((NN  + threads - 1) / threads, BC);
    dim3 gridPix  ((HWP + threads - 1) / threads, BC);

    if (ws_size >= need) {
        NodePack* pack = (NodePack*)d_ws;
        build_packed_kernel<<<gridNodes, threads, 0, stream>>>(attrs, thr, levels, parent, pack);
        walk_packed_kernel <<<gridPix,   threads, 0, stream>>>(pack, pix, out);
    } else {
        walk_direct_kernel <<<gridPix,   threads, 0, stream>>>(attrs, thr, levels, parent, pix, out);
    }
}